// bspline_64381559767111
// MI455X (gfx1250) — compile-verified
//
#include <hip/hip_runtime.h>

#define T_DIM 68      // knots
#define C_DIM 64      // coefficients / degree-3 basis count
#define NK    67      // degree-0 basis count (T_DIM-1)
#define NI    69      // elementary intervals (incl. two unbounded zero intervals)
#define NROWS 288     // NI*4 = 276 rows padded to 18 WMMA tiles of 16
#define NTILES (NROWS/16)

// d_ws layout (float offsets)
#define WS_SORT 0     // 68 sorted knots
#define WS_COEF 80    // NROWS floats: per-interval cubic coeffs (interval m -> [m*4 .. m*4+3]) ; 16B aligned
#define WS_P3   512   // NROWS x C_DIM floats (P3 basis-polynomial matrix, row-major)
// total: 512 + 288*64 = 18944 floats = 75,776 bytes of scratch

typedef float v2f __attribute__((ext_vector_type(2)));
typedef float v4f __attribute__((ext_vector_type(4)));
typedef float v8f __attribute__((ext_vector_type(8)));

// ---------------------------------------------------------------------------
// Kernel 1: sort knots + build per-interval cubic polynomials of every basis
// function (symbolic Cox-de Boor on degree<=3 polynomials in u = x - s_left).
// One block per padded interval row-group; entries across lanes, neighbor
// exchange through LDS, everything stays in registers.
// ---------------------------------------------------------------------------
__global__ __launch_bounds__(96) void bs_setup(const float* __restrict__ t,
                                               float* __restrict__ ws) {
  __shared__ float st[T_DIM];   // original knots
  __shared__ float ss[T_DIM];   // sorted knots
  __shared__ v4f   xch[NK];     // neighbor exchange of polynomial coeffs

  const int tid = threadIdx.x;
  const int m   = blockIdx.x;   // 0..71 (intervals 0..68 real, 69..71 pad)

  if (tid < T_DIM) st[tid] = t[tid];
  __syncthreads();
  if (tid < T_DIM) {
    const float v = st[tid];
    int r = 0;
    for (int j = 0; j < T_DIM; ++j) {
      const float u = st[j];
      r += (u < v) || (u == v && j < tid);   // stable rank sort
    }
    ss[r] = v;
  }
  __syncthreads();
  if (m == 0 && tid < T_DIM) ws[WS_SORT + tid] = ss[tid];

  const bool realiv = (m >= 1 && m <= 67);   // interval [ss[m-1], ss[m])
  const float x0 = realiv ? ss[m - 1] : 0.0f;

  const int i = tid;                         // basis-entry index
  v4f P = {0.0f, 0.0f, 0.0f, 0.0f};          // poly coeffs: P.x + P.y*u + P.z*u^2 + P.w*u^3
  if (i < NK) {
    // degree-0 indicator, constant over the half-open interval; evaluate at
    // left endpoint with the reference's exact comparisons.
    P.x = (realiv && (st[i] <= x0) && (x0 < st[i + 1])) ? 1.0f : 0.0f;
  }

  for (int k = 1; k <= 3; ++k) {
    __syncthreads();
    if (i < NK) xch[i] = P;
    __syncthreads();
    const int mm = NK - k;
    if (i < mm) {
      const v4f q = xch[i + 1];
      const float d1 = st[i + k] - st[i];
      const float w1 = (d1 != 0.0f) ? 1.0f / d1 : 0.0f;       // where(d1!=0, ., 0)
      const float d2 = st[i + k + 1] - st[i + 1];
      const float w2 = (d2 != 0.0f) ? 1.0f / d2 : 0.0f;
      // factor1 = (x - t_i)/d1     = a1*u + b1
      // factor2 = (t_{i+k+1}-x)/d2 = a2*u + b2
      const float a1 = w1,  b1 = w1 * (x0 - st[i]);
      const float a2 = -w2, b2 = w2 * (st[i + k + 1] - x0);
      v4f r;
      r.x = b1 * P.x + b2 * q.x;
      r.y = b1 * P.y + a1 * P.x + b2 * q.y + a2 * q.x;
      r.z = b1 * P.z + a1 * P.y + b2 * q.z + a2 * q.y;
      r.w = b1 * P.w + a1 * P.z + b2 * q.w + a2 * q.z;
      P = r;
    } else {
      P = (v4f){0.0f, 0.0f, 0.0f, 0.0f};
    }
  }

  // Write the four coefficient-rows of this interval: row = m*4 + d, K = basis i
  if (i < C_DIM) {
    const int base = WS_P3 + (m * 4) * C_DIM + i;
    ws[base + 0 * C_DIM] = P.x;
    ws[base + 1 * C_DIM] = P.y;
    ws[base + 2 * C_DIM] = P.z;
    ws[base + 3 * C_DIM] = P.w;
  }
}

// ---------------------------------------------------------------------------
// Kernel 2: coef[row] = sum_i c[i] * P3[row][i]  via V_WMMA_F32_16X16X4_F32.
// 18 waves, one 16-row tile each; K=64 accumulated in 16 chunks of 4.
// A (16x4 f32): lanes 0-15 hold rows with K={0,1} in VGPR{0,1}; lanes 16-31
// hold K={2,3}. B's columns are all equal to the c-chunk, so D[:,n] is the
// result for every n; extract N=0 from the documented 16x16 f32 C/D layout.
// ---------------------------------------------------------------------------
__global__ __launch_bounds__(32 * NTILES) void bs_combine(const float* __restrict__ cc,
                                                          float* __restrict__ ws) {
  const int lane = threadIdx.x & 31;
  const int tile = threadIdx.x >> 5;      // 0..17
  const int half = lane >> 4;             // 0: K=0,1   1: K=2,3
  const int l    = lane & 15;             // M row within tile
  const float* __restrict__ P3 = ws + WS_P3;
  const int row = tile * 16 + l;

  v8f acc = {};
  for (int kk = 0; kk < C_DIM; kk += 4) {
    v2f a = *(const v2f*)(P3 + row * C_DIM + kk + 2 * half);
    v2f b = *(const v2f*)(cc + kk + 2 * half);   // broadcast c-chunk into all columns
    acc = __builtin_amdgcn_wmma_f32_16x16x4_f32(
        /*neg_a=*/false, a, /*neg_b=*/false, b,
        /*c_mod=*/(short)0, acc, /*reuse_a=*/false, /*reuse_b=*/false);
  }
  // D VGPR v: lanes 0-15 -> (M=v, N=lane); lanes 16-31 -> (M=v+8, N=lane-16)
  if (l == 0) {
    const int base = tile * 16 + half * 8;
    #pragma unroll
    for (int v = 0; v < 8; ++v) ws[WS_COEF + base + v] = acc[v];
  }
}

// ---------------------------------------------------------------------------
// Kernel 3: streaming evaluation. Binary search pos = #(sorted knots <= x)
// (half-open intervals exactly as the reference), then Horner on the cubic
// in u = x - s_left. Memory-bound: one B128 load + one B128 store per 4 pts.
// ---------------------------------------------------------------------------
__device__ __forceinline__ float bs_eval1(float xs, const float* sknot, const v4f* scoef) {
  int pos = 0;
  #pragma unroll
  for (int step = 64; step >= 1; step >>= 1) {
    const int np = pos + step;
    int id = np - 1;
    id = id > (T_DIM - 1) ? (T_DIM - 1) : id;
    const bool ok = (np <= T_DIM) & (sknot[id] <= xs);   // branchless
    pos = ok ? np : pos;
  }
  const v4f cf = scoef[pos];                    // pos==0 / pos==68 -> zero poly
  const int pi = pos > 0 ? pos - 1 : 0;
  const float u = xs - sknot[pi];
  return ((cf.w * u + cf.z) * u + cf.y) * u + cf.x;
}

__global__ __launch_bounds__(256) void bs_eval(const float* __restrict__ x,
                                               const float* __restrict__ ws,
                                               float* __restrict__ out, int n) {
  __shared__ float sknot[T_DIM];
  __shared__ v4f   scoef[NI];
  const int tid = threadIdx.x;
  if (tid < T_DIM) sknot[tid] = ws[WS_SORT + tid];
  if (tid < NI)    scoef[tid] = *(const v4f*)(ws + WS_COEF + tid * 4);
  __syncthreads();

  const int gsz = gridDim.x * blockDim.x;
  const int gid = blockIdx.x * blockDim.x + tid;
  const int n4  = n >> 2;

  for (int i = gid; i < n4; i += gsz) {
    v4f xv = __builtin_nontemporal_load((const v4f*)x + i);
    v4f r;
    #pragma unroll
    for (int e = 0; e < 4; ++e) r[e] = bs_eval1(xv[e], sknot, scoef);
    __builtin_nontemporal_store(r, (v4f*)out + i);
  }
  for (int i = (n4 << 2) + gid; i < n; i += gsz) {   // tail (N%4 != 0 safety)
    out[i] = bs_eval1(x[i], sknot, scoef);
  }
}

// ---------------------------------------------------------------------------
extern "C" void kernel_launch(void* const* d_in, const int* in_sizes, int n_in,
                              void* d_out, int out_size, void* d_ws, size_t ws_size,
                              hipStream_t stream) {
  const float* x = (const float*)d_in[0];
  const float* t = (const float*)d_in[1];
  const float* c = (const float*)d_in[2];
  float* out = (float*)d_out;
  float* ws  = (float*)d_ws;
  const int n = in_sizes[0];

  bs_setup<<<NROWS / 4, 96, 0, stream>>>(t, ws);          // 72 blocks
  bs_combine<<<1, 32 * NTILES, 0, stream>>>(c, ws);       // 18 waves, WMMA mat-vec

  const int n4 = n >> 2;
  int blocks = (n4 + 255) / 256;
  if (blocks > 2048) blocks = 2048;
  if (blocks < 1) blocks = 1;
  bs_eval<<<blocks, 256, 0, stream>>>(x, ws, out, n);
}